// ACRGNN_66855460929770
// MI455X (gfx1250) — compile-verified
//
#include <hip/hip_runtime.h>
#include <hip/hip_bf16.h>

typedef __attribute__((ext_vector_type(2))) float v2f;
typedef __attribute__((ext_vector_type(8))) float v8f;

#define NODES   50000
#define EDGES   800000
#define GRAPHS  64
#define IN_DIM  64
#define HID     128
#define LAYERS  3
#define BN_EPS  1e-5f

// ---------------- pad: h[n,0:64]=x, h[n,64:128]=0 ----------------
__global__ __launch_bounds__(256) void pad_kernel(const float* __restrict__ x,
                                                  float* __restrict__ h, int n) {
    int t = blockIdx.x * blockDim.x + threadIdx.x;      // n*32 items, 4 floats each
    int node = t >> 5;
    int c = (t & 31) << 2;
    if (node >= n) return;
    float4 v;
    if (c < IN_DIM) v = *(const float4*)(x + (size_t)node * IN_DIM + c);
    else            v = make_float4(0.f, 0.f, 0.f, 0.f);
    *(float4*)(h + (size_t)node * HID + c) = v;
}

// ---------------- edge scatter: aggr[dst] += h[src] ----------------
__global__ __launch_bounds__(256) void edge_scatter(const float* __restrict__ h,
                                                    const long long* __restrict__ ei,
                                                    float* __restrict__ aggr, int nEdges) {
    long long t = (long long)blockIdx.x * blockDim.x + threadIdx.x;
    int e = (int)(t >> 5);
    int c = (int)(t & 31) << 2;
    if (e >= nEdges) return;
    long long s = ei[e];
    long long d = ei[(long long)nEdges + e];
    float4 v = *(const float4*)(h + (size_t)s * HID + c);
    float* dp = aggr + (size_t)d * HID + c;
    atomicAdd(dp + 0, v.x);
    atomicAdd(dp + 1, v.y);
    atomicAdd(dp + 2, v.z);
    atomicAdd(dp + 3, v.w);
}

// ---------------- readout: gsum[batch[n]] += h[n] ----------------
__global__ __launch_bounds__(256) void node_readout(const float* __restrict__ h,
                                                    const long long* __restrict__ batch,
                                                    float* __restrict__ gsum, int n) {
    long long t = (long long)blockIdx.x * blockDim.x + threadIdx.x;
    int node = (int)(t >> 5);
    int c = (int)(t & 31) << 2;
    if (node >= n) return;
    int g = (int)batch[node];
    float4 v = *(const float4*)(h + (size_t)node * HID + c);
    float* dp = gsum + (size_t)g * HID + c;
    atomicAdd(dp + 0, v.x);
    atomicAdd(dp + 1, v.y);
    atomicAdd(dp + 2, v.z);
    atomicAdd(dp + 3, v.w);
}

// ---------------- fused WMMA GEMM: tmp = relu(h@V^T + aggr@A^T + gsum[batch]@R^T + biases)
// grid.x = NODES/16 blocks, 256 threads = 8 waves, wave w -> feature cols [16w,16w+16)
__global__ __launch_bounds__(256) void gemm_fused(const float* __restrict__ h,
                                                  const float* __restrict__ aggr,
                                                  const float* __restrict__ gsum,
                                                  const long long* __restrict__ batch,
                                                  const float* __restrict__ Vw,
                                                  const float* __restrict__ Vb,
                                                  const float* __restrict__ Aw,
                                                  const float* __restrict__ Ab,
                                                  const float* __restrict__ Rw,
                                                  const float* __restrict__ Rb,
                                                  float* __restrict__ out) {
    const int lane = threadIdx.x & 31;
    const int wave = threadIdx.x >> 5;
    const int row0 = blockIdx.x << 4;          // node tile base
    const int col0 = wave << 4;                // feature tile base
    const int m    = lane & 15;                // A row index / B col index
    const int kb   = (lane >> 4) << 1;         // 0 (lanes 0-15) or 2 (lanes 16-31)
    const int row  = row0 + m;

    // A-operand row pointers for the three products
    const float* a0 = h    + (size_t)row * HID;
    const float* a1 = aggr + (size_t)row * HID;
    const float* a2 = gsum + (size_t)batch[row] * HID;
    // B-operand: B[k][n] = W[col0+n][k], W row-major [HID][HID]
    const float* w0 = Vw + (size_t)(col0 + m) * HID;
    const float* w1 = Aw + (size_t)(col0 + m) * HID;
    const float* w2 = Rw + (size_t)(col0 + m) * HID;

    v8f acc = {};
    const float* ap[3] = {a0, a1, a2};
    const float* wp[3] = {w0, w1, w2};
#pragma unroll
    for (int mtx = 0; mtx < 3; ++mtx) {
        const float* A = ap[mtx];
        const float* W = wp[mtx];
#pragma unroll
        for (int k = 0; k < HID; k += 4) {
            v2f a; a.x = A[k + kb]; a.y = A[k + kb + 1];
            v2f b; b.x = W[k + kb]; b.y = W[k + kb + 1];
            // D = A(16x4) x B(4x16) + C ; fp32 exact path
            acc = __builtin_amdgcn_wmma_f32_16x16x4_f32(
                /*neg_a=*/false, a, /*neg_b=*/false, b,
                /*c_mod=*/(short)0, acc, /*reuse_a=*/false, /*reuse_b=*/false);
        }
    }

    // bias + relu + store. C layout: VGPR r -> M = r + (lane>=16 ? 8 : 0), N = lane&15
    const int j = col0 + m;
    const float bsum = Vb[j] + Ab[j] + Rb[j];
    const int moff = (lane >> 4) << 3;
#pragma unroll
    for (int r = 0; r < 8; ++r) {
        float v = acc[r] + bsum;
        v = v > 0.f ? v : 0.f;
        out[(size_t)(row0 + moff + r) * HID + j] = v;
    }
}

// ---------------- BN stats: stats[f] += sum, stats[128+f] += sumsq ----------------
#define BN_NPB 256   // nodes per block
__global__ __launch_bounds__(256) void bn_stats(const float* __restrict__ t,
                                                float* __restrict__ stats, int n) {
    const int f    = threadIdx.x & (HID - 1);
    const int half = threadIdx.x >> 7;          // 0 or 1
    const int base = blockIdx.x * BN_NPB;
    const int end  = (base + BN_NPB < n) ? (base + BN_NPB) : n;
    float s = 0.f, ss = 0.f;
    for (int r = base + half; r < end; r += 2) {
        float v = t[(size_t)r * HID + f];
        s += v; ss += v * v;
    }
    __shared__ float ls[256], lss[256];
    ls[threadIdx.x] = s; lss[threadIdx.x] = ss;
    __syncthreads();
    if (half == 0) {
        atomicAdd(&stats[f],       s  + ls[threadIdx.x + 128]);
        atomicAdd(&stats[HID + f], ss + lss[threadIdx.x + 128]);
    }
}

// ---------------- BN apply: h = (t-mean)*rsqrt(var+eps)*gamma + beta ----------------
__global__ __launch_bounds__(256) void bn_apply(const float* __restrict__ t,
                                                const float* __restrict__ stats,
                                                const float* __restrict__ gamma,
                                                const float* __restrict__ beta,
                                                float* __restrict__ h, int n) {
    long long idx = (long long)blockIdx.x * blockDim.x + threadIdx.x;
    if (idx >= (long long)n * HID) return;
    int f = (int)(idx & (HID - 1));
    const float invN = 1.0f / (float)n;
    float mean = stats[f] * invN;
    float var  = stats[HID + f] * invN - mean * mean;
    float sc   = __frsqrt_rn(var + BN_EPS) * gamma[f];
    h[idx] = (t[idx] - mean) * sc + beta[f];
}

// ---------------- final projection: out[n,0:2] = h[n] @ lin_w^T + lin_b ----------------
__global__ __launch_bounds__(256) void final_lin(const float* __restrict__ h,
                                                 const float* __restrict__ lw,
                                                 const float* __restrict__ lb,
                                                 float* __restrict__ out, int n) {
    int node = blockIdx.x * blockDim.x + threadIdx.x;
    if (node >= n) return;
    float a0 = lb[0], a1 = lb[1];
    const float* hp = h + (size_t)node * HID;
#pragma unroll 8
    for (int k = 0; k < HID; ++k) {
        float v = hp[k];
        a0 = fmaf(v, lw[k],       a0);
        a1 = fmaf(v, lw[HID + k], a1);
    }
    out[(size_t)node * 2 + 0] = a0;
    out[(size_t)node * 2 + 1] = a1;
}

extern "C" void kernel_launch(void* const* d_in, const int* in_sizes, int n_in,
                              void* d_out, int out_size, void* d_ws, size_t ws_size,
                              hipStream_t stream) {
    const float*     x     = (const float*)d_in[0];
    const long long* ei    = (const long long*)d_in[1];
    const long long* batch = (const long long*)d_in[2];
    const float*     V_w   = (const float*)d_in[3];
    const float*     V_b   = (const float*)d_in[4];
    const float*     A_w   = (const float*)d_in[5];
    const float*     A_b   = (const float*)d_in[6];
    const float*     R_w   = (const float*)d_in[7];
    const float*     R_b   = (const float*)d_in[8];
    const float*     gam   = (const float*)d_in[9];
    const float*     bet   = (const float*)d_in[10];
    const float*     lw    = (const float*)d_in[11];
    const float*     lb    = (const float*)d_in[12];
    float*           out   = (float*)d_out;

    const int n = in_sizes[0] / IN_DIM;   // 50000
    const int e = in_sizes[1] / 2;        // 800000

    // workspace carve (floats)
    float* hbuf  = (float*)d_ws;                        // n*HID
    float* tmp   = hbuf  + (size_t)n * HID;             // n*HID
    float* aggr  = tmp   + (size_t)n * HID;             // n*HID
    float* gsum  = aggr  + (size_t)n * HID;             // GRAPHS*HID
    float* stats = gsum  + (size_t)GRAPHS * HID;        // 2*HID

    // pad input to 128 features
    {
        long long items = (long long)n * 32;
        pad_kernel<<<(unsigned)((items + 255) / 256), 256, 0, stream>>>(x, hbuf, n);
    }

    for (int l = 0; l < LAYERS; ++l) {
        hipMemsetAsync(aggr,  0, (size_t)n * HID * sizeof(float), stream);
        hipMemsetAsync(gsum,  0, (size_t)GRAPHS * HID * sizeof(float), stream);
        hipMemsetAsync(stats, 0, 2 * HID * sizeof(float), stream);

        {
            long long items = (long long)e * 32;
            edge_scatter<<<(unsigned)((items + 255) / 256), 256, 0, stream>>>(hbuf, ei, aggr, e);
        }
        {
            long long items = (long long)n * 32;
            node_readout<<<(unsigned)((items + 255) / 256), 256, 0, stream>>>(hbuf, batch, gsum, n);
        }
        gemm_fused<<<n / 16, 256, 0, stream>>>(
            hbuf, aggr, gsum, batch,
            V_w + (size_t)l * HID * HID, V_b + (size_t)l * HID,
            A_w + (size_t)l * HID * HID, A_b + (size_t)l * HID,
            R_w + (size_t)l * HID * HID, R_b + (size_t)l * HID,
            tmp);
        bn_stats<<<(n + BN_NPB - 1) / BN_NPB, 256, 0, stream>>>(tmp, stats, n);
        {
            long long items = (long long)n * HID;
            bn_apply<<<(unsigned)((items + 255) / 256), 256, 0, stream>>>(
                tmp, stats, gam + (size_t)l * HID, bet + (size_t)l * HID, hbuf, n);
        }
    }

    final_lin<<<(n + 255) / 256, 256, 0, stream>>>(hbuf, lw, lb, out, n);
}